// Net_68590627717528
// MI455X (gfx1250) — compile-verified
//
#include <hip/hip_runtime.h>
#include <hip/hip_bf16.h>

#define IDIM 80
#define ODIM 80
#define HDIM 1024
#define CDIM 64
#define NTOK 4096          // B*T = 4*1024
#define NSHIFT 159         // 2*ODIM-1 windows
#define PAD 79             // ODIM-1

typedef float v2f __attribute__((ext_vector_type(2)));
typedef float v8f __attribute__((ext_vector_type(8)));

// ---------------------------------------------------------------------------
// Stage 1: per-token alignment search + softmax gating -> x_ele (NTOK x 80)
// ---------------------------------------------------------------------------
__global__ __launch_bounds__(256) void stage1(const float* __restrict__ x,
                                              const float* __restrict__ y,
                                              float* __restrict__ xele) {
    const int token = blockIdx.x;
    const int tid   = threadIdx.x;

    __shared__ float xs[IDIM];
    __shared__ float ys[ODIM];
    __shared__ float yaas[ODIM];
    __shared__ float red[256];
    __shared__ int   redi[256];

    if (tid < IDIM) { xs[tid] = x[token * IDIM + tid]; ys[tid] = y[token * ODIM + tid]; }
    __syncthreads();

    // ||y||
    red[tid] = (tid < ODIM) ? ys[tid] * ys[tid] : 0.0f;
    __syncthreads();
    for (int s = 128; s > 0; s >>= 1) { if (tid < s) red[tid] += red[tid + s]; __syncthreads(); }
    const float ny = sqrtf(red[0]);
    __syncthreads();

    // cosine similarity for each of the 159 shifts; argmax with first-max tie rule
    float sim = -3.402823466e38f;
    if (tid < NSHIFT) {
        float dot = 0.f, nx = 0.f;
        int jlo = PAD - tid; if (jlo < 0) jlo = 0;
        int jhi = (2 * PAD) - tid; if (jhi > IDIM - 1) jhi = IDIM - 1;
        for (int j = jlo; j <= jhi; ++j) {
            float xv = xs[tid + j - PAD];
            dot += ys[j] * xv;
            nx  += xv * xv;
        }
        sim = dot / (ny * sqrtf(nx) + 1e-6f);
        if (sim != sim) sim = 0.f;   // NaN -> 0 like reference
    }
    red[tid] = sim; redi[tid] = tid;
    __syncthreads();
    for (int s = 128; s > 0; s >>= 1) {
        if (tid < s) {
            float o = red[tid + s]; int oi = redi[tid + s];
            if (o > red[tid] || (o == red[tid] && oi < redi[tid])) { red[tid] = o; redi[tid] = oi; }
        }
        __syncthreads();
    }
    const int theta = redi[0];
    __syncthreads();

    // y_align = window at theta; d2 = ||y_align||*||y|| + eps
    float ya = 0.f;
    if (tid < ODIM) {
        int src = theta - PAD + tid;
        if (src >= 0 && src < IDIM) ya = xs[src];
    }
    red[tid] = (tid < ODIM) ? ya * ya : 0.f;
    __syncthreads();
    for (int s = 128; s > 0; s >>= 1) { if (tid < s) red[tid] += red[tid + s]; __syncthreads(); }
    const float d2 = sqrtf(red[0]) * ny + 1e-6f;
    __syncthreads();

    // softmax(score / TEMPER)
    float sc = 0.f;
    if (tid < ODIM) sc = (ya * ys[tid] / d2) * 0.1f;   // score / 10.0
    red[tid] = (tid < ODIM) ? sc : -3.402823466e38f;
    __syncthreads();
    for (int s = 128; s > 0; s >>= 1) {
        if (tid < s) { if (red[tid + s] > red[tid]) red[tid] = red[tid + s]; }
        __syncthreads();
    }
    const float smax = red[0];
    __syncthreads();
    float e = (tid < ODIM) ? expf(sc - smax) : 0.f;
    red[tid] = e;
    __syncthreads();
    for (int s = 128; s > 0; s >>= 1) { if (tid < s) red[tid] += red[tid + s]; __syncthreads(); }
    const float ssum = red[0];
    __syncthreads();

    if (tid < ODIM) {
        float attn = e / ssum;
        if (attn != attn) attn = 0.f;
        yaas[tid] = ya * attn;
    }
    __syncthreads();

    // un-shift: x_ele[j] = yaa[j + PAD - theta] (zero outside)
    if (tid < ODIM) {
        int src = tid + PAD - theta;
        float v = (src >= 0 && src < ODIM) ? yaas[src] : 0.f;
        xele[token * ODIM + tid] = v;
    }
}

// ---------------------------------------------------------------------------
// GEMM1: h = x_ele @ W_enc + b_enc   (M=4096, N=1024, K=80)  via WMMA f32
// One wave computes one 16x16 output tile; 20 x v_wmma_f32_16x16x4_f32.
// ---------------------------------------------------------------------------
__global__ __launch_bounds__(128) void gemm_enc(const float* __restrict__ xele,
                                                const float* __restrict__ W,
                                                const float* __restrict__ bias,
                                                float* __restrict__ h) {
    const int wave = threadIdx.x >> 5;
    const int lane = threadIdx.x & 31;
    const int tile = blockIdx.x * 4 + wave;          // 16384 tiles total
    const int mt = tile >> 6;                        // 256 M-tiles
    const int nt = tile & 63;                        // 64  N-tiles
    const int l15   = lane & 15;
    const int lhi   = lane >> 4;                     // 0|1
    const int m     = (mt << 4) + l15;               // A: row = lane%16
    const int n     = (nt << 4) + l15;               // B/D: col = lane%16
    const int khalf = lhi << 1;                      // K offset 0|2

    v8f c = {};
    #pragma unroll 4
    for (int k0 = 0; k0 < IDIM; k0 += 4) {
        const int kb = k0 + khalf;                   // K = k0 + v + 2*(lane>=16)
        v2f a, b;
        a.x = xele[m * IDIM + kb];
        a.y = xele[m * IDIM + kb + 1];
        b.x = W[kb * HDIM + n];
        b.y = W[(kb + 1) * HDIM + n];
        c = __builtin_amdgcn_wmma_f32_16x16x4_f32(false, a, false, b, (short)0, c,
                                                  false, false);
    }
    const float bn = bias[n];
    const int mrow = (mt << 4) + (lhi << 3);         // D: M = r + 8*(lane>=16)
    #pragma unroll
    for (int r = 0; r < 8; ++r)
        h[(mrow + r) * HDIM + n] = c[r] + bn;
}

// ---------------------------------------------------------------------------
// Top-64 mask on h*h (exact top_k semantics incl. index-ordered ties)
// Bit patterns of non-negative floats are monotone as uint32 -> binary search
// the 64th-largest pattern, then keep strictly-greater + first ties.
// ---------------------------------------------------------------------------
__global__ __launch_bounds__(256) void topk_mask(float* __restrict__ h) {
    const int token = blockIdx.x;
    float* hr = h + token * HDIM;
    const int tid = threadIdx.x;

    __shared__ unsigned int hv[HDIM];
    __shared__ unsigned char keep[HDIM];
    __shared__ int cnt;

    for (int i = tid; i < HDIM; i += 256) {
        float v = hr[i];
        hv[i] = __float_as_uint(v * v);   // >= 0 -> uint order == float order
        keep[i] = 0;
    }
    __syncthreads();

    // largest c with count(v >= c) >= CDIM  ==  CDIM-th largest value
    unsigned int lo = 0u, hi = 0xFFFFFFFFu;
    while (lo < hi) {
        unsigned int mid = lo + ((hi - lo + 1u) >> 1);
        if (tid == 0) cnt = 0;
        __syncthreads();
        int c = 0;
        for (int i = tid; i < HDIM; i += 256) if (hv[i] >= mid) ++c;
        atomicAdd(&cnt, c);
        __syncthreads();
        int total = cnt;
        __syncthreads();
        if (total >= CDIM) lo = mid; else hi = mid - 1u;
    }
    const unsigned int V = lo;

    // keep strictly greater
    if (tid == 0) cnt = 0;
    __syncthreads();
    {
        int c = 0;
        for (int i = tid; i < HDIM; i += 256)
            if (hv[i] > V) { keep[i] = 1; ++c; }
        atomicAdd(&cnt, c);
    }
    __syncthreads();
    // first (CDIM - cntGT) ties in index order (jax top_k tie rule)
    if (tid == 0) {
        int rem = CDIM - cnt;
        for (int i = 0; i < HDIM && rem > 0; ++i)
            if (hv[i] == V) { keep[i] = 1; --rem; }
    }
    __syncthreads();

    for (int i = tid; i < HDIM; i += 256)
        if (!keep[i]) hr[i] = 0.f;
}

// ---------------------------------------------------------------------------
// GEMM2: out = h @ W_dec + b_dec   (M=4096, N=80, K=1024) via WMMA f32
// ---------------------------------------------------------------------------
__global__ __launch_bounds__(128) void gemm_dec(const float* __restrict__ h,
                                                const float* __restrict__ W,
                                                const float* __restrict__ bias,
                                                float* __restrict__ out) {
    const int wave = threadIdx.x >> 5;
    const int lane = threadIdx.x & 31;
    const int tile = blockIdx.x * 4 + wave;          // 1280 tiles total
    const int mt = tile / 5;                         // 256 M-tiles
    const int nt = tile % 5;                         // 5   N-tiles
    const int l15   = lane & 15;
    const int lhi   = lane >> 4;
    const int m     = (mt << 4) + l15;
    const int n     = (nt << 4) + l15;
    const int khalf = lhi << 1;

    v8f c = {};
    #pragma unroll 8
    for (int k0 = 0; k0 < HDIM; k0 += 4) {
        const int kb = k0 + khalf;
        v2f a, b;
        a.x = h[m * HDIM + kb];
        a.y = h[m * HDIM + kb + 1];
        b.x = W[kb * ODIM + n];
        b.y = W[(kb + 1) * ODIM + n];
        c = __builtin_amdgcn_wmma_f32_16x16x4_f32(false, a, false, b, (short)0, c,
                                                  false, false);
    }
    const float bn = bias[n];
    const int mrow = (mt << 4) + (lhi << 3);
    #pragma unroll
    for (int r = 0; r < 8; ++r)
        out[(mrow + r) * ODIM + n] = c[r] + bn;
}

// ---------------------------------------------------------------------------
extern "C" void kernel_launch(void* const* d_in, const int* in_sizes, int n_in,
                              void* d_out, int out_size, void* d_ws, size_t ws_size,
                              hipStream_t stream) {
    const float* x     = (const float*)d_in[0];
    const float* y     = (const float*)d_in[1];
    const float* W_enc = (const float*)d_in[2];
    const float* b_enc = (const float*)d_in[3];
    const float* W_dec = (const float*)d_in[4];
    const float* b_dec = (const float*)d_in[5];
    float* out = (float*)d_out;

    float* xele = (float*)d_ws;                 // NTOK * 80 floats
    float* hbuf = xele + (size_t)NTOK * ODIM;   // NTOK * 1024 floats

    stage1   <<<NTOK, 256, 0, stream>>>(x, y, xele);
    gemm_enc <<<(NTOK / 16) * (HDIM / 16) / 4, 128, 0, stream>>>(xele, W_enc, b_enc, hbuf);
    topk_mask<<<NTOK, 256, 0, stream>>>(hbuf);
    gemm_dec <<<(NTOK / 16) * (ODIM / 16) / 4, 128, 0, stream>>>(hbuf, W_dec, b_dec, out);
}